// GATv2Layer4View_21878563406398
// MI455X (gfx1250) — compile-verified
//
#include <hip/hip_runtime.h>

typedef float v2f __attribute__((ext_vector_type(2)));
typedef float v8f __attribute__((ext_vector_type(8)));

namespace {
constexpr int kB = 4, kV = 4, kN = 1000, kInF = 64, kHeads = 4, kOutF = 32;
constexpr int kD = 128;               // HEADS*OUT_F
constexpr int kE = 64000;
constexpr int kNtot = kB * kN;        // 4000
constexpr int kEtot = kE + kNtot;     // 68000 (with self loops)
constexpr int kRows = kB * kV * kN;   // 16000 rows (both x-rows and MHA seq rows)
constexpr float kNegSlope = 0.2f;
}

// ---- monotone float <-> uint encoding for atomicMax on floats ----
__device__ __forceinline__ unsigned f2u_ord(float f) {
  unsigned u = __float_as_uint(f);
  return (u & 0x80000000u) ? ~u : (u | 0x80000000u);
}
__device__ __forceinline__ float u2f_ord(unsigned u) {
  return __uint_as_float((u & 0x80000000u) ? (u ^ 0x80000000u) : ~u);
}

// seq-flat row r (= (b*N+n)*V + v) -> row index in [B,V,N,*] storage
__device__ __forceinline__ int seq_to_bvn_row(int r) {
  int m = r >> 2;            // V == 4
  int v = r & 3;
  int b = m / kN;
  int n = m - b * kN;
  return (b * kV + v) * kN + n;
}

// =====================  1) h = x @ W^T  (f32 WMMA)  =====================
// x: [16000, 64] rows in [B,V,N] order; W: [128, 64]; h: [16000, 128]
// Each wave: one 16-col N-tile x two 16-row M-tiles (B fragment reused).
__global__ __launch_bounds__(32) void k_proj(const float* __restrict__ x,
                                             const float* __restrict__ W,
                                             float* __restrict__ h) {
  int tile = blockIdx.x;         // 500 M-blocks * 8 N-tiles
  int mb = tile >> 3;
  int nt = tile & 7;
  int lane = threadIdx.x & 31;
  int half = lane >> 4;          // 0: K pair {0,1}, 1: K pair {2,3} (per ISA 16x4 layout)
  int l15  = lane & 15;
  int m0 = mb * 32, n0 = nt * 16;
  const float* arow0 = x + (size_t)(m0 + l15) * kInF;
  const float* arow1 = x + (size_t)(m0 + 16 + l15) * kInF;
  const float* brow  = W + (size_t)(n0 + l15) * kInF;   // B[k][n] = W[n][k]
  v8f acc0 = {}, acc1 = {};
  for (int k0 = 0; k0 < kInF; k0 += 4) {
    int ka = k0 + 2 * half;
    v2f a0 = *(const v2f*)(arow0 + ka);
    v2f a1 = *(const v2f*)(arow1 + ka);
    v2f b  = *(const v2f*)(brow + ka);
    acc0 = __builtin_amdgcn_wmma_f32_16x16x4_f32(false, a0, false, b, (short)0, acc0,
                                                 false, false);
    acc1 = __builtin_amdgcn_wmma_f32_16x16x4_f32(false, a1, false, b, (short)0, acc1,
                                                 false, false);
  }
  int col = n0 + l15;
  for (int i = 0; i < 8; ++i) {
    int row = m0 + i + (half ? 8 : 0);
    h[(size_t)row * kD + col] = acc0[i];
    h[(size_t)(row + 16) * kD + col] = acc1[i];
  }
}

// =====================  2) edge scores + segment max  =====================
// score[e,h] = sum_f att[h,f]*lrelu(h0[src][h,f]) + att[h,32+f]*lrelu(h0[dst][h,f])
// h0[n] = h row n (the b0,v0 block) if n < 1000, else zeros.
__global__ void k_scores(const float* __restrict__ h,
                         const float* __restrict__ att,
                         const int* __restrict__ edge,      // [2, E]
                         float* __restrict__ score,
                         unsigned* __restrict__ segmax) {
  int t = blockIdx.x * blockDim.x + threadIdx.x;
  if (t >= kEtot * kHeads) return;
  int e  = t >> 2;
  int hd = t & 3;
  int s, d;
  if (e < kE) { s = edge[e]; d = edge[kE + e]; }
  else        { s = e - kE;  d = s; }          // self loops
  const float* ah = att + hd * 2 * kOutF;
  float sc = 0.f;
  if (s < kN) {
    const float* hs = h + (size_t)s * kD + hd * kOutF;
    for (int f = 0; f < kOutF; ++f) {
      float v = hs[f];
      v = v > 0.f ? v : v * kNegSlope;
      sc += ah[f] * v;
    }
  }
  if (d < kN) {
    const float* hp = h + (size_t)d * kD + hd * kOutF;
    for (int f = 0; f < kOutF; ++f) {
      float v = hp[f];
      v = v > 0.f ? v : v * kNegSlope;
      sc += ah[kOutF + f] * v;
    }
  }
  score[t] = sc;
  atomicMax(&segmax[d * kHeads + hd], f2u_ord(sc));
}

// =====================  3) exp + denominator  =====================
__global__ void k_exp(const int* __restrict__ edge,
                      const float* __restrict__ score,
                      const unsigned* __restrict__ segmax,
                      float* __restrict__ ex,
                      float* __restrict__ denom) {
  int t = blockIdx.x * blockDim.x + threadIdx.x;
  if (t >= kEtot * kHeads) return;
  int e  = t >> 2;
  int hd = t & 3;
  int d = (e < kE) ? edge[kE + e] : (e - kE);
  float m = u2f_ord(segmax[d * kHeads + hd]);
  float ev = expf(score[t] - m);
  ex[t] = ev;
  atomicAdd(&denom[d * kHeads + hd], ev);
}

// =====================  4) scatter aggregation  =====================
// Only edges with src and dst in the same node block contribute; broadcast over V views.
__global__ void k_agg(const int* __restrict__ edge,
                      const float* __restrict__ ex,
                      const float* __restrict__ denom,
                      const float* __restrict__ h,
                      float* __restrict__ hgat) {
  int t = blockIdx.x * blockDim.x + threadIdx.x;
  if (t >= kEtot * kD) return;
  int e = t >> 7;            // D == 128
  int f = t & 127;
  int s, d;
  if (e < kE) { s = edge[e]; d = edge[kE + e]; }
  else        { s = e - kE;  d = s; }
  int bs = s / kN, bd = d / kN;
  if (bs != bd) return;
  int hd = f >> 5;
  float alpha = ex[e * kHeads + hd] / (denom[d * kHeads + hd] + 1e-16f);
  int ns = s - bs * kN, nd = d - bd * kN;
  for (int v = 0; v < kV; ++v) {
    size_t srow = (size_t)((bd * kV + v) * kN + ns) * kD + f;
    size_t drow = (size_t)((bd * kV + v) * kN + nd) * kD + f;
    atomicAdd(&hgat[drow], alpha * h[srow]);
  }
}

// =====================  5) qkv = seq @ in_proj_w^T + b  (f32 WMMA)  =====================
// seq row r gathers from hgat[B,V,N,128]; qkv: [16000, 384] in seq-row order.
__global__ __launch_bounds__(32) void k_qkv(const float* __restrict__ hg,
                                            const float* __restrict__ ipw,  // [384,128]
                                            const float* __restrict__ ipb,  // [384]
                                            float* __restrict__ qkv) {
  int tile = blockIdx.x;           // 500 M-blocks * 24 N-tiles
  int mb = tile / 24;
  int nt = tile - mb * 24;
  int lane = threadIdx.x & 31;
  int half = lane >> 4, l15 = lane & 15;
  int m0 = mb * 32, n0 = nt * 16;
  const float* arow0 = hg + (size_t)seq_to_bvn_row(m0 + l15) * kD;
  const float* arow1 = hg + (size_t)seq_to_bvn_row(m0 + 16 + l15) * kD;
  const float* brow  = ipw + (size_t)(n0 + l15) * kD;     // B[k][n] = ipw[n][k]
  v8f acc0 = {}, acc1 = {};
  for (int k0 = 0; k0 < kD; k0 += 4) {
    int ka = k0 + 2 * half;
    v2f a0 = *(const v2f*)(arow0 + ka);
    v2f a1 = *(const v2f*)(arow1 + ka);
    v2f b  = *(const v2f*)(brow + ka);
    acc0 = __builtin_amdgcn_wmma_f32_16x16x4_f32(false, a0, false, b, (short)0, acc0,
                                                 false, false);
    acc1 = __builtin_amdgcn_wmma_f32_16x16x4_f32(false, a1, false, b, (short)0, acc1,
                                                 false, false);
  }
  int col = n0 + l15;
  float bv = ipb[col];
  for (int i = 0; i < 8; ++i) {
    int row = m0 + i + (half ? 8 : 0);
    qkv[(size_t)row * 384 + col] = acc0[i] + bv;
    qkv[(size_t)(row + 16) * 384 + col] = acc1[i] + bv;
  }
}

// =====================  6) tiny MHA over the V axis  =====================
// one wave per (node m, head); lanes carry the hd=32 dim.
__global__ __launch_bounds__(128) void k_attn(const float* __restrict__ qkv,
                                              float* __restrict__ ao) {
  int m    = blockIdx.x;           // 0..3999
  int head = threadIdx.x >> 5;     // 4 waves = 4 heads
  int lane = threadIdx.x & 31;
  const float scale = 0.17677669529663687f;   // 1/sqrt(32)
  float q[4], k[4], v[4];
  for (int vi = 0; vi < 4; ++vi) {
    const float* row = qkv + (size_t)(m * 4 + vi) * 384;
    q[vi] = row[head * 32 + lane];
    k[vi] = row[128 + head * 32 + lane];
    v[vi] = row[256 + head * 32 + lane];
  }
  float aw[4][4];
  for (int qi = 0; qi < 4; ++qi)
    for (int ki = 0; ki < 4; ++ki) {
      float p = q[qi] * k[ki];
      for (int off = 16; off; off >>= 1) p += __shfl_xor(p, off, 32);
      aw[qi][ki] = p * scale;
    }
  for (int qi = 0; qi < 4; ++qi) {
    float mx = aw[qi][0];
    for (int ki = 1; ki < 4; ++ki) mx = fmaxf(mx, aw[qi][ki]);
    float sum = 0.f;
    for (int ki = 0; ki < 4; ++ki) { aw[qi][ki] = expf(aw[qi][ki] - mx); sum += aw[qi][ki]; }
    float inv = 1.f / sum;
    float o = 0.f;
    for (int ki = 0; ki < 4; ++ki) o += aw[qi][ki] * inv * v[ki];
    ao[(size_t)(m * 4 + qi) * kD + head * 32 + lane] = o;
  }
}

// =====================  7) out = ao @ out_proj_w^T + biases (f32 WMMA)  =====================
// scatter stores to output layout [B,V,N,D].
__global__ __launch_bounds__(32) void k_out(const float* __restrict__ ao,
                                            const float* __restrict__ opw,  // [128,128]
                                            const float* __restrict__ opb,
                                            const float* __restrict__ bias,
                                            float* __restrict__ out) {
  int tile = blockIdx.x;           // 500 M-blocks * 8 N-tiles
  int mb = tile >> 3, nt = tile & 7;
  int lane = threadIdx.x & 31;
  int half = lane >> 4, l15 = lane & 15;
  int m0 = mb * 32, n0 = nt * 16;
  const float* arow0 = ao + (size_t)(m0 + l15) * kD;
  const float* arow1 = ao + (size_t)(m0 + 16 + l15) * kD;
  const float* brow  = opw + (size_t)(n0 + l15) * kD;
  v8f acc0 = {}, acc1 = {};
  for (int k0 = 0; k0 < kD; k0 += 4) {
    int ka = k0 + 2 * half;
    v2f a0 = *(const v2f*)(arow0 + ka);
    v2f a1 = *(const v2f*)(arow1 + ka);
    v2f b  = *(const v2f*)(brow + ka);
    acc0 = __builtin_amdgcn_wmma_f32_16x16x4_f32(false, a0, false, b, (short)0, acc0,
                                                 false, false);
    acc1 = __builtin_amdgcn_wmma_f32_16x16x4_f32(false, a1, false, b, (short)0, acc1,
                                                 false, false);
  }
  int col = n0 + l15;
  float add = opb[col] + bias[col];
  for (int i = 0; i < 8; ++i) {
    int row = m0 + i + (half ? 8 : 0);
    out[(size_t)seq_to_bvn_row(row) * kD + col] = acc0[i] + add;
    out[(size_t)seq_to_bvn_row(row + 16) * kD + col] = acc1[i] + add;
  }
}

extern "C" void kernel_launch(void* const* d_in, const int* in_sizes, int n_in,
                              void* d_out, int out_size, void* d_ws, size_t ws_size,
                              hipStream_t stream) {
  const float* x    = (const float*)d_in[0];   // [B,V,N,64]
  const float* W    = (const float*)d_in[1];   // [128,64]
  const float* att  = (const float*)d_in[2];   // [1,4,64]
  const float* ipw  = (const float*)d_in[3];   // [384,128]
  const float* ipb  = (const float*)d_in[4];   // [384]
  const float* opw  = (const float*)d_in[5];   // [128,128]
  const float* opb  = (const float*)d_in[6];   // [128]
  const float* bias = (const float*)d_in[7];   // [128]
  const int*   edge = (const int*)d_in[8];     // [2,E]
  float* out = (float*)d_out;

  float* ws    = (float*)d_ws;
  float* h     = ws;                                    // 16000*128
  float* hgat  = h     + (size_t)kRows * kD;            // 16000*128
  float* qkv   = hgat  + (size_t)kRows * kD;            // 16000*384
  float* ao    = qkv   + (size_t)kRows * 384;           // 16000*128
  float* score = ao    + (size_t)kRows * kD;            // 68000*4
  float* ex    = score + (size_t)kEtot * kHeads;        // 68000*4
  float* denom = ex    + (size_t)kEtot * kHeads;        // 4000*4
  unsigned* smax = (unsigned*)(denom + (size_t)kNtot * kHeads);  // 4000*4

  hipMemsetAsync(hgat, 0, (size_t)kRows * kD * sizeof(float), stream);
  hipMemsetAsync(denom, 0, (size_t)kNtot * kHeads * sizeof(float), stream);
  hipMemsetAsync(smax, 0, (size_t)kNtot * kHeads * sizeof(unsigned), stream);

  k_proj<<<500 * 8, 32, 0, stream>>>(x, W, h);
  int nt1 = kEtot * kHeads;
  k_scores<<<(nt1 + 255) / 256, 256, 0, stream>>>(h, att, edge, score, smax);
  k_exp<<<(nt1 + 255) / 256, 256, 0, stream>>>(edge, score, smax, ex, denom);
  int nt2 = kEtot * kD;
  k_agg<<<(nt2 + 255) / 256, 256, 0, stream>>>(edge, ex, denom, h, hgat);
  k_qkv<<<500 * 24, 32, 0, stream>>>(hgat, ipw, ipb, qkv);
  k_attn<<<kNtot, 128, 0, stream>>>(qkv, ao);
  k_out<<<500 * 8, 32, 0, stream>>>(ao, opw, opb, bias, out);
}